// ContrastiveLoss_80633716015714
// MI455X (gfx1250) — compile-verified
//
#include <hip/hip_runtime.h>
#include <math.h>

#define NUMC    19
#define MCOMP   5
#define KTOT    95
#define KPAD    96
#define AF      64
#define NPIX    262144      // 4*256*256
#define IGN     255
#define LOG2PI  1.8378770664093453f
#define TEMP_INV 0.01f

typedef float v2f __attribute__((ext_vector_type(2)));
typedef float v8f __attribute__((ext_vector_type(8)));
typedef int   v4i __attribute__((ext_vector_type(4)));

__device__ __forceinline__ v8f wmma4(v2f a, v2f b, v8f c) {
  // D = A(16x4 f32) * B(4x16 f32) + C(16x16 f32)
  return __builtin_amdgcn_wmma_f32_16x16x4_f32(false, a, false, b, (short)0, c,
                                               false, false);
}

// ---- CDNA5 async-to-LDS path (guarded; falls back to sync staging) --------
#if defined(__AMDGCN__) && __has_builtin(__builtin_amdgcn_global_load_async_to_lds_b128)
#define HAVE_ASYNC 1
typedef __attribute__((address_space(1))) v4i* gv4p;
typedef __attribute__((address_space(3))) v4i* lv4p;
#else
#define HAVE_ASYNC 0
#endif

#if defined(__AMDGCN__) && __has_builtin(__builtin_amdgcn_s_wait_asynccnt)
#define WAIT_ASYNC(n) __builtin_amdgcn_s_wait_asynccnt(n)
#elif defined(__AMDGCN__)
#define WAIT_ASYNC(n) asm volatile("s_wait_asynccnt " #n ::: "memory")
#else
#define WAIT_ASYNC(n)
#endif

// ---------------------------------------------------------------- init / fin
__global__ void k_init(float* acc) {
  if (threadIdx.x < 2) acc[threadIdx.x] = 0.f;
}

__global__ void k_fin(const float* __restrict__ acc, float* __restrict__ out) {
  out[0] = acc[0] / fmaxf(acc[1], 1.f);
}

// ------------------------------------------------------------- precompute W
__global__ __launch_bounds__(128) void k_pre(
    const float* __restrict__ mean, const float* __restrict__ cov,
    const float* __restrict__ tm, const float* __restrict__ ctg,
    const float* __restrict__ csr, float* __restrict__ w1p,
    float* __restrict__ w2, float* __restrict__ locW, float* __restrict__ mt,
    float* __restrict__ cfix, float* __restrict__ ratio) {
  int t = threadIdx.x;
  if (t < KTOT) {
    float q = 0.f, ld = 0.f;
    for (int a = 0; a < AF; ++a) {
      float l = mean[t * AF + a];
      float s = cov[t * AF + a];
      float iv = 1.f / (s * s);
      w2[t * AF + a] = iv;
      w1p[t * AF + a] = -2.f * l * iv;   // fold the -2 of the cross term
      locW[t * AF + a] = l;
      q += l * l * iv;
      ld += __logf(s);
    }
    cfix[t] = -0.5f * ((float)AF * LOG2PI + 2.f * ld + q);
  } else if (t == KTOT) {               // zero pad row 95
    for (int a = 0; a < AF; ++a) {
      w2[t * AF + a] = 0.f; w1p[t * AF + a] = 0.f; locW[t * AF + a] = 0.f;
    }
    cfix[t] = 0.f;
  }
  if (t < NUMC) {
    // target_memory[c,a,:100] mean then l2-normalize over a
    float ss = 0.f;
    for (int a = 0; a < AF; ++a) {
      const float* row = tm + ((size_t)t * AF + a) * 100;
      float s = 0.f;
      for (int j = 0; j < 100; ++j) s += row[j];
      s *= 0.01f;
      ss += s * s;
    }
    float rn = 1.f / fmaxf(sqrtf(ss), 1e-12f);
    for (int a = 0; a < AF; ++a) {
      const float* row = tm + ((size_t)t * AF + a) * 100;
      float s = 0.f;
      for (int j = 0; j < 100; ++j) s += row[j];
      mt[t * AF + a] = s * 0.01f * rn;
    }
    float r = ctg[t] / csr[t];          // nan_to_num semantics
    if (__builtin_isnan(r)) r = 0.f;
    else if (__builtin_isinf(r)) r = copysignf(3.4028234663852886e38f, r);
    ratio[t] = r;
  } else if (t < 32) {                  // zero pad mt rows 19..31
    for (int a = 0; a < AF; ++a) mt[t * AF + a] = 0.f;
  }
}

// -------------------------------------------------------- label downscale 4x4
__global__ __launch_bounds__(256) void k_labels(const int* __restrict__ mask,
                                                int* __restrict__ mlab) {
  int p = blockIdx.x * 256 + threadIdx.x;      // over B*256*256
  int b = p >> 16;
  int hw = p & 65535;
  int h = hw >> 8, w = hw & 255;
  const int4* mp = (const int4*)(mask + (size_t)b * (1024 * 1024) +
                                 (size_t)(h * 4) * 1024 + w * 4);
  int lab[16];
#pragma unroll
  for (int i = 0; i < 4; ++i) {
    int4 v = mp[i * 256];                // row stride 1024 ints = 256 int4
    lab[i * 4 + 0] = v.x; lab[i * 4 + 1] = v.y;
    lab[i * 4 + 2] = v.z; lab[i * 4 + 3] = v.w;
  }
  int best = 0, bestc = -1;
#pragma unroll
  for (int c = 0; c <= NUMC; ++c) {            // class 19 == mapped IGNORE
    int tgt = (c == NUMC) ? IGN : c;
    int cnt = 0;
#pragma unroll
    for (int q = 0; q < 16; ++q) cnt += (lab[q] == tgt);
    if (cnt > bestc) { bestc = cnt; best = c; }   // first-max over class order
  }
  // ratio < 0.75  <=>  count < 12 (exact: 0.75*16 == 12)
  mlab[p] = (best == NUMC || bestc < 12) ? IGN : best;
}

// ------------------------------------------------------------- fused main
#define TILES 4
#define XS2  68    // x staging stride ([channel][pixel]), 16B-aligned rows
#define XS   66    // W LDS row stride
#define MS   100
#define S4S  36

#define OFF_X0   0
#define OFF_X1   4352
#define OFF_W1   8704
#define OFF_W2   15040
#define OFF_LOC  21376
#define OFF_MT   27712
#define OFF_MAHA 29824
#define OFF_S3   36224
#define OFF_S4   42624
#define OFF_RED  44928
#define OFF_CFX  44930
#define OFF_RAT  45026
#define SMEM_TOT 45056     // 180,224 B < 320KB WGP LDS

__device__ __forceinline__ void stage_tile(float* xb,
                                           const float* __restrict__ feat,
                                           int tbase, int wv, int ln) {
  // stage 64 pixels x 64 channels as xb[channel][pixel]; wave wv owns 8 chans
  int b = tbase >> 16;
  int hw0 = tbase & 65535;
  int half = ln >> 4;
  int pixg = (ln & 15) * 4;
#pragma unroll
  for (int i = 0; i < 4; ++i) {
    int ch = wv * 8 + i * 2 + half;
    const float* gp = feat + ((size_t)b << 22) + ((size_t)ch << 16) + hw0 + pixg;
    float* lp = xb + ch * XS2 + pixg;
#if HAVE_ASYNC
    __builtin_amdgcn_global_load_async_to_lds_b128((gv4p)gp, (lv4p)lp, 0, 0);
#else
    float4 v = *(const float4*)gp;
    *(float4*)lp = v;
#endif
  }
}

__global__ __launch_bounds__(256, 1) void k_main(
    const float* __restrict__ feat, const int* __restrict__ mlab,
    const float* __restrict__ w1p, const float* __restrict__ w2g,
    const float* __restrict__ locW, const float* __restrict__ mtg,
    const float* __restrict__ cfix, const float* __restrict__ ratio,
    float* __restrict__ acc) {
  __shared__ float sm[SMEM_TOT];
  float* w1l  = sm + OFF_W1;
  float* w2l  = sm + OFF_W2;
  float* locl = sm + OFF_LOC;
  float* mtl  = sm + OFF_MT;
  float* mahal= sm + OFF_MAHA;
  float* s3l  = sm + OFF_S3;
  float* s4l  = sm + OFF_S4;
  float* cfxl = sm + OFF_CFX;
  float* ratl = sm + OFF_RAT;

  int t = threadIdx.x;
  int wv = t >> 5, ln = t & 31;
  int nl = ln & 15, hi = ln >> 4;                   // lane -> (M/N, K-half)

  if (t == 0) { sm[OFF_RED] = 0.f; sm[OFF_RED + 1] = 0.f; }
  if (t < KPAD) cfxl[t] = cfix[t];
  if (t < NUMC) ratl[t] = ratio[t];

  // stage weight matrices once per block (L2-resident)
  for (int i = t; i < KPAD * AF; i += 256) {
    int r = i >> 6, c = i & 63;
    w1l[r * XS + c]  = w1p[i];
    w2l[r * XS + c]  = w2g[i];
    locl[r * XS + c] = locW[i];
  }
  for (int i = t; i < 32 * AF; i += 256) {
    int r = i >> 6, c = i & 63;
    mtl[r * XS + c] = mtg[i];
  }

  int base = blockIdx.x * (TILES * 64);
  stage_tile(sm + OFF_X0, feat, base, wv, ln);       // prologue: tile 0

  for (int ti = 0; ti < TILES; ++ti) {
    float* xb = sm + ((ti & 1) ? OFF_X1 : OFF_X0);
    if (ti + 1 < TILES) {                            // prefetch next tile
      stage_tile(sm + (((ti + 1) & 1) ? OFF_X1 : OFF_X0), feat,
                 base + (ti + 1) * 64, wv, ln);
      WAIT_ASYNC(4);                                 // current tile done
    } else {
      WAIT_ASYNC(0);
    }
    __syncthreads();

    if (t < 64) {                                    // l2-normalize pixel cols
      float ss = 0.f;
#pragma unroll
      for (int a = 0; a < AF; ++a) { float v = xb[a * XS2 + t]; ss += v * v; }
      float rn = 1.f / fmaxf(sqrtf(ss), 1e-12f);
#pragma unroll
      for (int a = 0; a < AF; ++a) xb[a * XS2 + t] *= rn;
    }
    __syncthreads();

    // --------------- WMMA jobs: 24 maha + 24 logits + 8 sim = 56 -----------
    for (int job = wv; job < 56; job += 8) {
      v8f c = {};
      if (job < 24) {                  // maha = x^2 @ iv + x @ (-2*loc*iv)
        int rb = job / 6, ctl = job % 6;
        int r0 = rb * 16, n0 = ctl * 16;
        const float* xcol  = xb + r0 + nl + hi * 2 * XS2;
        const float* brow2 = w2l + (n0 + nl) * XS + hi * 2;
        const float* brow1 = w1l + (n0 + nl) * XS + hi * 2;
#pragma unroll
        for (int kk = 0; kk < 16; ++kk) {
          v2f av, bv;
          av.x = xcol[kk * 4 * XS2]; av.y = xcol[kk * 4 * XS2 + XS2];
          av.x *= av.x; av.y *= av.y;               // x^2 on the fly
          bv.x = brow2[kk * 4]; bv.y = brow2[kk * 4 + 1];
          c = wmma4(av, bv, c);
        }
#pragma unroll
        for (int kk = 0; kk < 16; ++kk) {
          v2f av, bv;
          av.x = xcol[kk * 4 * XS2]; av.y = xcol[kk * 4 * XS2 + XS2];
          bv.x = brow1[kk * 4]; bv.y = brow1[kk * 4 + 1];
          c = wmma4(av, bv, c);                     // accumulate into same C
        }
        float* orow = mahal + (r0 + hi * 8) * MS + n0 + nl;
#pragma unroll
        for (int i = 0; i < 8; ++i) orow[i * MS] = c[i];
      } else if (job < 48) {                        // logits = x @ loc^T
        int j = job - 24;
        int rb = j / 6, ctl = j % 6;
        int r0 = rb * 16, n0 = ctl * 16;
        const float* xcol = xb + r0 + nl + hi * 2 * XS2;
        const float* brow = locl + (n0 + nl) * XS + hi * 2;
#pragma unroll
        for (int kk = 0; kk < 16; ++kk) {
          v2f av, bv;
          av.x = xcol[kk * 4 * XS2]; av.y = xcol[kk * 4 * XS2 + XS2];
          bv.x = brow[kk * 4]; bv.y = brow[kk * 4 + 1];
          c = wmma4(av, bv, c);
        }
        float* orow = s3l + (r0 + hi * 8) * MS + n0 + nl;
#pragma unroll
        for (int i = 0; i < 8; ++i) orow[i * MS] = c[i];
      } else {                                      // sim = x @ mean_target^T
        int j = job - 48;
        int rb = j >> 1, ctl = j & 1;
        int r0 = rb * 16, n0 = ctl * 16;
        const float* xcol = xb + r0 + nl + hi * 2 * XS2;
        const float* brow = mtl + (n0 + nl) * XS + hi * 2;
#pragma unroll
        for (int kk = 0; kk < 16; ++kk) {
          v2f av, bv;
          av.x = xcol[kk * 4 * XS2]; av.y = xcol[kk * 4 * XS2 + XS2];
          bv.x = brow[kk * 4]; bv.y = brow[kk * 4 + 1];
          c = wmma4(av, bv, c);
        }
        float* orow = s4l + (r0 + hi * 8) * S4S + n0 + nl;
#pragma unroll
        for (int i = 0; i < 8; ++i) orow[i * S4S] = c[i];
      }
    }
    __syncthreads();

    // --------------- per-pixel softmax / argmax / CE ------------------------
    if (t < 64) {
      const float* mrow = mahal + t * MS;
      const float* srow = s3l + t * MS;
      const float* s4r  = s4l + t * S4S;
      float mx4 = -3.4e38f;
#pragma unroll
      for (int cc = 0; cc < NUMC; ++cc) mx4 = fmaxf(mx4, s4r[cc]);
      float mx95 = -3.4e38f;
#pragma unroll 5
      for (int k = 0; k < KTOT; ++k)
        mx95 = fmaxf(mx95, fmaf(-0.5f, mrow[k], cfxl[k]));
      // argmax_c ratio_c * exp(s4_c - mx4) * sum_m exp(lp - mx95)
      // (class-common positive normalizers dropped: argmax invariant)
      float bestsc = -1.f; int label = 0;
      for (int cc = 0; cc < NUMC; ++cc) {
        float ssum = 0.f;
#pragma unroll
        for (int m = 0; m < MCOMP; ++m) {
          float lp = fmaf(-0.5f, mrow[cc * 5 + m], cfxl[cc * 5 + m]);
          ssum += __expf(lp - mx95);
        }
        float sc = ratl[cc] * __expf(s4r[cc] - mx4) * ssum;
        if (sc > bestsc) { bestsc = sc; label = cc; }
      }
      // cross-entropy on max-over-M logits / TEMP
      float mxl = -3.4e38f, lmL = 0.f;
      for (int cc = 0; cc < NUMC; ++cc) {
        float lm = -3.4e38f;
#pragma unroll
        for (int m = 0; m < MCOMP; ++m) lm = fmaxf(lm, srow[cc * 5 + m]);
        lm *= TEMP_INV;
        mxl = fmaxf(mxl, lm);
        if (cc == label) lmL = lm;
      }
      float z = 0.f;
      for (int cc = 0; cc < NUMC; ++cc) {
        float lm = -3.4e38f;
#pragma unroll
        for (int m = 0; m < MCOMP; ++m) lm = fmaxf(lm, srow[cc * 5 + m]);
        z += __expf(lm * TEMP_INV - mxl);
      }
      float ce = -(lmL - mxl - __logf(z));
      if (mlab[base + ti * 64 + t] != IGN) {
        atomicAdd(&sm[OFF_RED], ce);
        atomicAdd(&sm[OFF_RED + 1], 1.f);
      }
    }
    __syncthreads();
  }

  if (t == 0) {
    atomicAdd(&acc[0], sm[OFF_RED]);
    atomicAdd(&acc[1], sm[OFF_RED + 1]);
  }
}

// ---------------------------------------------------------------- launcher
extern "C" void kernel_launch(void* const* d_in, const int* in_sizes, int n_in,
                              void* d_out, int out_size, void* d_ws,
                              size_t ws_size, hipStream_t stream) {
  (void)in_sizes; (void)n_in; (void)out_size; (void)ws_size;
  const float* feat = (const float*)d_in[0];
  const int*   mask = (const int*)d_in[1];
  const float* mean = (const float*)d_in[2];
  const float* cov  = (const float*)d_in[3];
  const float* tm   = (const float*)d_in[4];
  const float* ctg  = (const float*)d_in[5];
  const float* csr  = (const float*)d_in[6];

  char* ws   = (char*)d_ws;
  float* acc  = (float*)ws;                         // 2 floats
  float* w1p  = (float*)(ws + 256);                 // [96][64]
  float* w2   = (float*)(ws + 256 + 24576);         // [96][64]
  float* locW = (float*)(ws + 256 + 2 * 24576);     // [96][64]
  float* mt   = (float*)(ws + 256 + 3 * 24576);     // [32][64]
  float* cfix = (float*)(ws + 82176);               // [96]
  float* ratio= (float*)(ws + 82688);               // [19]
  int*   mlab = (int*)(ws + 83200);                 // [NPIX]

  k_init<<<1, 32, 0, stream>>>(acc);
  k_pre<<<1, 128, 0, stream>>>(mean, cov, tm, ctg, csr, w1p, w2, locW, mt,
                               cfix, ratio);
  k_labels<<<NPIX / 256, 256, 0, stream>>>(mask, mlab);
  k_main<<<NPIX / 256, 256, 0, stream>>>(feat, mlab, w1p, w2, locW, mt, cfix,
                                         ratio, acc);
  k_fin<<<1, 1, 0, stream>>>(acc, (float*)d_out);
}